// GraphConvLayer_33956011442633
// MI455X (gfx1250) — compile-verified
//
#include <hip/hip_runtime.h>

typedef __attribute__((ext_vector_type(2))) float v2f;
typedef __attribute__((ext_vector_type(4))) float v4f;
typedef __attribute__((ext_vector_type(8))) float v8f;

#define N_NODES 50000
#define N_EDGES 800000
#define D_IN    128
#define D_OUT   128

// ---------------------------------------------------------------------------
// Kernel 1: h = x @ w  (fp32, V_WMMA_F32_16X16X4_F32)
// Block = 256 threads = 8 waves. Block b owns rows [16b, 16b+16).
// Wave w owns the 16x16 output tile at columns [16w, 16w+16).
// A tile (16x4 f32): lane m=lane&15 is the row; VGPR j holds K = k + 2*(lane>>4) + j.
// B tile (4x16 f32): lane n=lane&15 is the col; VGPR j holds K = k + 2*(lane>>4) + j.
// D (v8f): VGPR i -> row r0 + i + 8*(lane>>4), col c0 + (lane&15).
// ---------------------------------------------------------------------------
__global__ __launch_bounds__(256) void gemm_xw_wmma(const float* __restrict__ x,
                                                    const float* __restrict__ w,
                                                    float* __restrict__ h) {
    const int wave = threadIdx.x >> 5;   // 0..7 : column tile
    const int lane = threadIdx.x & 31;
    const int r0 = blockIdx.x * 16;      // 50000/16 = 3125 row tiles exactly
    const int c0 = wave * 16;

    const int m  = lane & 15;            // row (A) / col (B) within tile
    const int kh = lane >> 4;            // which K pair this half-wave holds

    v8f acc = {};
    const float* ap_base = x + (size_t)(r0 + m) * D_IN + 2 * kh;
    const float* bp_base = w + (size_t)(2 * kh) * D_OUT + c0 + m;

    for (int k = 0; k < D_IN; k += 4) {
        v2f a;
        const float* ap = ap_base + k;          // two consecutive K values
        a.x = ap[0];
        a.y = ap[1];

        v2f b;
        const float* bp = bp_base + (size_t)k * D_OUT;  // two rows of w, stride D_OUT
        b.x = bp[0];
        b.y = bp[D_OUT];

        // 8 args: (neg_a, A, neg_b, B, c_mod, C, reuse_a, reuse_b)
        acc = __builtin_amdgcn_wmma_f32_16x16x4_f32(
            /*neg_a=*/false, a, /*neg_b=*/false, b,
            /*c_mod=*/(short)0, acc, /*reuse_a=*/false, /*reuse_b=*/false);
    }

    float* dp = h + (size_t)(r0 + 8 * kh) * D_OUT + c0 + m;
#pragma unroll
    for (int i = 0; i < 8; ++i)
        dp[(size_t)i * D_OUT] = acc[i];
}

// ---------------------------------------------------------------------------
// Kernel 2: CSR row pointers from sorted COO rows (lower_bound per node).
// rowptr has N_NODES+1 entries.
// ---------------------------------------------------------------------------
__global__ __launch_bounds__(256) void build_rowptr(const int* __restrict__ rows,
                                                    int* __restrict__ rowptr) {
    const int i = blockIdx.x * blockDim.x + threadIdx.x;
    if (i > N_NODES) return;
    int lo = 0, hi = N_EDGES;
    while (lo < hi) {
        const int mid = (lo + hi) >> 1;
        if (rows[mid] < i) lo = mid + 1; else hi = mid;
    }
    rowptr[i] = lo;
}

// ---------------------------------------------------------------------------
// Kernel 3: out[r] = relu( sum_e vals[e] * h[cols[e]] ), one wave32 per node.
// Lane owns 4 contiguous output columns (float4). No atomics: rows are sorted,
// segment bounds come from rowptr. Gathers of h are served out of L2
// (h = 25.6 MB << 192 MB L2).
// ---------------------------------------------------------------------------
__global__ __launch_bounds__(256) void spmm_relu(const float* __restrict__ h,
                                                 const int* __restrict__ cols,
                                                 const float* __restrict__ vals,
                                                 const int* __restrict__ rowptr,
                                                 float* __restrict__ out) {
    const int wave = threadIdx.x >> 5;
    const int lane = threadIdx.x & 31;
    const int node = blockIdx.x * 8 + wave;
    if (node >= N_NODES) return;

    const int beg = rowptr[node];
    const int end = rowptr[node + 1];

    const v4f* __restrict__ hv = (const v4f*)h;  // 32 v4f per row (D_OUT/4)
    v4f acc = {};
    for (int e = beg; e < end; ++e) {
        const int   c = cols[e];   // wave-uniform -> scalarized load
        const float v = vals[e];
        const v4f row = hv[(size_t)c * (D_OUT / 4) + lane];
        acc.x += v * row.x;
        acc.y += v * row.y;
        acc.z += v * row.z;
        acc.w += v * row.w;
    }
    acc.x = acc.x > 0.0f ? acc.x : 0.0f;
    acc.y = acc.y > 0.0f ? acc.y : 0.0f;
    acc.z = acc.z > 0.0f ? acc.z : 0.0f;
    acc.w = acc.w > 0.0f ? acc.w : 0.0f;

    ((v4f*)out)[(size_t)node * (D_OUT / 4) + lane] = acc;
}

// ---------------------------------------------------------------------------
extern "C" void kernel_launch(void* const* d_in, const int* in_sizes, int n_in,
                              void* d_out, int out_size, void* d_ws, size_t ws_size,
                              hipStream_t stream) {
    const float* x        = (const float*)d_in[0];
    const float* w        = (const float*)d_in[1];
    const int*   adj_rows = (const int*)d_in[2];
    const int*   adj_cols = (const int*)d_in[3];
    const float* adj_vals = (const float*)d_in[4];
    float*       out      = (float*)d_out;

    // workspace layout: [ h : N_NODES*D_OUT f32 | rowptr : N_NODES+1 i32 ]
    float* h      = (float*)d_ws;
    int*   rowptr = (int*)((char*)d_ws + (size_t)N_NODES * D_OUT * sizeof(float));

    gemm_xw_wmma<<<N_NODES / 16, 256, 0, stream>>>(x, w, h);
    build_rowptr<<<(N_NODES + 1 + 255) / 256, 256, 0, stream>>>(adj_rows, rowptr);
    spmm_relu<<<(N_NODES + 7) / 8, 256, 0, stream>>>(h, adj_cols, adj_vals, rowptr, out);
}